// GNN_80530636800660
// MI455X (gfx1250) — compile-verified
//
#include <hip/hip_runtime.h>
#include <hip/hip_bf16.h>

// ---------------------------------------------------------------------------
// GNN (SAGE + SAGPooling cascade + attention pooling) for MI455X / gfx1250.
// Heavy GEMM (x @ [Wl;Wr]^T, K=1036) runs on V_WMMA_F32_16X16X4_F32.
// ---------------------------------------------------------------------------

typedef __attribute__((ext_vector_type(2))) float v2f;
typedef __attribute__((ext_vector_type(8))) float v8f;

#define BATCH 64
#define NNODE 512
#define NEDGE 4096
#define XDIM  1036
#define HDIM  128
#define FDIM  32
#define NPAT  16
#define NEGINF (-3.402823466e38f)

// ---------------- workspace layout (float offsets) -------------------------
static constexpr size_t OFF_COLINV = 0;                                  // 16
static constexpr size_t OFF_WCAT   = 16;                                 // 256*1036
static constexpr size_t OFF_PQ     = OFF_WCAT + 256ull * XDIM;           // 64*512*256
static constexpr size_t OFF_PQ2    = OFF_PQ + (size_t)BATCH * NNODE * 256; // 64*128*256
static constexpr size_t OFF_XB     = OFF_PQ2 + (size_t)BATCH * 128 * 256;  // 64*128*128
static constexpr size_t OFF_ROUT   = OFF_XB + (size_t)BATCH * 128 * 128;   // 64*256

// ---------------------------------------------------------------------------
// Kernel A: per-column max over x[:, :, c] for c < 12 -> colinv[c] = 1/max
// ---------------------------------------------------------------------------
__global__ __launch_bounds__(256) void gnn_colmax(const float* __restrict__ x,
                                                  float* __restrict__ colinv) {
  __shared__ float red[256];
  const int c   = blockIdx.x;      // 0..11
  const int tid = threadIdx.x;
  float m = NEGINF;
  for (size_t i = tid; i < (size_t)BATCH * NNODE; i += 256)
    m = fmaxf(m, x[i * XDIM + c]);
  red[tid] = m;
  __syncthreads();
  for (int off = 128; off; off >>= 1) {
    if (tid < off) red[tid] = fmaxf(red[tid], red[tid + off]);
    __syncthreads();
  }
  if (tid == 0) colinv[c] = 1.0f / red[0];
}

// ---------------------------------------------------------------------------
// Kernel B: Wcat[256][1036] = [Wl ; Wr] with cols 0..11 scaled by colinv
// (folds the input normalization into the layer-0 weights)
// ---------------------------------------------------------------------------
__global__ __launch_bounds__(256) void gnn_wcat(const float* __restrict__ Wl,
                                                const float* __restrict__ Wr,
                                                const float* __restrict__ colinv,
                                                float* __restrict__ Wcat) {
  const int idx = blockIdx.x * 256 + threadIdx.x;
  if (idx >= 256 * XDIM) return;
  const int j = idx / XDIM;
  const int c = idx - j * XDIM;
  const float s = (c < 12) ? colinv[c] : 1.0f;
  const float w = (j < 128) ? Wl[(size_t)j * XDIM + c]
                            : Wr[(size_t)(j - 128) * XDIM + c];
  Wcat[idx] = w * s;
}

// ---------------------------------------------------------------------------
// Kernel C: PQ[g][i][j] = sum_k x[g][i][k] * Wcat[j][k]   (f32 WMMA 16x16x4)
// One wave -> one 16x64 output strip; A fragment reused across 4 WMMAs.
// Grid sized exactly: no divergence, EXEC all-ones (WMMA requirement).
// ---------------------------------------------------------------------------
__global__ __launch_bounds__(256) void gnn_gemm_wmma(const float* __restrict__ x,
                                                     const float* __restrict__ Wcat,
                                                     float* __restrict__ PQ) {
  const int lane = threadIdx.x & 31;
  const int wid  = blockIdx.x * 8 + (threadIdx.x >> 5);
  // 64 graphs * 32 Mtiles * 4 Nstrips = 8192 wave tasks
  const int g  = wid >> 7;
  const int r  = wid & 127;
  const int mt = r >> 2;
  const int ns = r & 3;
  const int Mbase = mt * 16;
  const int Nbase = ns * 64;
  const int l15   = lane & 15;
  const int koff  = (lane < 16) ? 0 : 2;   // ISA A/B fragment K split at lane 16

  const float* Arow = x + ((size_t)g * NNODE + Mbase + l15) * XDIM + koff;
  const float* B0 = Wcat + (size_t)(Nbase +  0 + l15) * XDIM + koff;
  const float* B1 = Wcat + (size_t)(Nbase + 16 + l15) * XDIM + koff;
  const float* B2 = Wcat + (size_t)(Nbase + 32 + l15) * XDIM + koff;
  const float* B3 = Wcat + (size_t)(Nbase + 48 + l15) * XDIM + koff;

  v8f a0 = {0.f,0.f,0.f,0.f,0.f,0.f,0.f,0.f};
  v8f a1 = a0, a2 = a0, a3 = a0;

  for (int k0 = 0; k0 < XDIM; k0 += 4) {
    v2f a = *(const v2f*)(Arow + k0);
    v2f b0 = *(const v2f*)(B0 + k0);
    v2f b1 = *(const v2f*)(B1 + k0);
    v2f b2 = *(const v2f*)(B2 + k0);
    v2f b3 = *(const v2f*)(B3 + k0);
    a0 = __builtin_amdgcn_wmma_f32_16x16x4_f32(false, a, false, b0, (short)0, a0, false, false);
    a1 = __builtin_amdgcn_wmma_f32_16x16x4_f32(false, a, false, b1, (short)0, a1, false, false);
    a2 = __builtin_amdgcn_wmma_f32_16x16x4_f32(false, a, false, b2, (short)0, a2, false, false);
    a3 = __builtin_amdgcn_wmma_f32_16x16x4_f32(false, a, false, b3, (short)0, a3, false, false);
  }

  // D layout: VGPR r -> M = Mbase + r + (lane<16 ? 0 : 8), N = Nbase + (lane&15)
  const int mofs = (lane < 16) ? 0 : 8;
  float* out = PQ + ((size_t)g * NNODE + Mbase + mofs) * 256 + Nbase + l15;
#pragma unroll
  for (int rr = 0; rr < 8; rr++) {
    out[(size_t)rr * 256 +  0] = a0[rr];
    out[(size_t)rr * 256 + 16] = a1[rr];
    out[(size_t)rr * 256 + 32] = a2[rr];
    out[(size_t)rr * 256 + 48] = a3[rr];
  }
}

// ---------------------------------------------------------------------------
// Kernel D: per-graph cascade (3x SAGE + score + top-k pool + readout)
// One block per graph; edges/masks in LDS; deterministic ballot-scan agg.
// ---------------------------------------------------------------------------
struct DP {
  const int* ei;          // [B,2,E]
  float* PQ;              // [B,512,256]  (cols 128..255 overwritten by h)
  float* PQ2;             // [B,128,256]
  float* xb;              // [B,128,128]
  float* routs;           // [B,256]
  const float* bl0;
  const float* Wl1; const float* bl1; const float* Wr1;
  const float* Wl2; const float* bl2; const float* Wr2;
  const float* Wrel0; const float* brel0; const float* Wroot0;
  const float* Wrel1; const float* brel1; const float* Wroot1;
  const float* Wrel2; const float* brel2; const float* Wroot2;
};

__global__ __launch_bounds__(256) void gnn_graph(DP p) {
  __shared__ int   s_src[NEDGE];
  __shared__ int   s_dst[NEDGE];
  __shared__ float s_mask[NEDGE];
  __shared__ int   s_cnt[NNODE];
  __shared__ int   s_nm[NNODE];
  __shared__ float s_score[NNODE];
  __shared__ float s_work[NNODE];
  __shared__ float s_t[NNODE];
  __shared__ float s_red[256];
  __shared__ int   s_redi[256];
  __shared__ int   s_perm[128];

  const int g    = blockIdx.x;
  const int tid  = threadIdx.x;
  const int lane = tid & 31;
  const int warp = tid >> 5;

  const int* eg = p.ei + (size_t)g * 2 * NEDGE;
  for (int e = tid; e < NEDGE; e += 256) {
    s_src[e]  = eg[e];
    s_dst[e]  = eg[NEDGE + e];
    s_mask[e] = 1.0f;
  }
  float* PQg  = p.PQ  + (size_t)g * NNODE * 256;
  float* PQ2g = p.PQ2 + (size_t)g * 128 * 256;
  float* xbg  = p.xb  + (size_t)g * 128 * 128;
  float* rg   = p.routs + (size_t)g * 256;
  for (int i = tid; i < 256; i += 256) rg[i] = 0.0f;

  int n = NNODE;
  const int ksz[3] = {103, 21, 5};

  for (int l = 0; l < 3; l++) {
    float* Pb = (l == 0) ? PQg : PQ2g;     // [n][256]: P cols 0..127, Q/h cols 128..255
    const float* bl    = (l == 0) ? p.bl0   : (l == 1 ? p.bl1   : p.bl2);
    const float* Wrel  = (l == 0) ? p.Wrel0 : (l == 1 ? p.Wrel1 : p.Wrel2);
    const float* Wroot = (l == 0) ? p.Wroot0: (l == 1 ? p.Wroot1: p.Wroot2);
    const float  brel  = ((l == 0) ? p.brel0 : (l == 1 ? p.brel1 : p.brel2))[0];
    const int k = ksz[l];

    __syncthreads();
    for (int i = tid; i < n; i += 256) s_cnt[i] = 0;
    __syncthreads();
    for (int e = tid; e < NEDGE; e += 256)
      if (s_mask[e] > 0.5f) atomicAdd(&s_cnt[s_dst[e]], 1);
    __syncthreads();

    // ---- SAGE: h[i] = relu(mean_neighbors(P) + bl + Q[i]); h written over Q
    for (int i = warp; i < n; i += 8) {
      float a0 = 0.f, a1 = 0.f, a2 = 0.f, a3 = 0.f;
      for (int eb = 0; eb < NEDGE; eb += 32) {
        const int e = eb + lane;
        const bool m = (s_dst[e] == i) && (s_mask[e] > 0.5f);
        unsigned long long bal = __ballot(m);
        while (bal) {
          const int j = __ffsll((long long)bal) - 1;
          bal &= bal - 1;
          const float* Pr = Pb + (size_t)s_src[eb + j] * 256;
          a0 += Pr[lane];      a1 += Pr[lane + 32];
          a2 += Pr[lane + 64]; a3 += Pr[lane + 96];
        }
      }
      const int   cn  = s_cnt[i];
      const float inv = 1.0f / (float)(cn > 1 ? cn : 1);
      float* Hr = Pb + (size_t)i * 256 + 128;
      const float q0 = Hr[lane],      q1 = Hr[lane + 32];
      const float q2 = Hr[lane + 64], q3 = Hr[lane + 96];
      Hr[lane]      = fmaxf(a0 * inv + bl[lane]      + q0, 0.0f);
      Hr[lane + 32] = fmaxf(a1 * inv + bl[lane + 32] + q1, 0.0f);
      Hr[lane + 64] = fmaxf(a2 * inv + bl[lane + 64] + q2, 0.0f);
      Hr[lane + 96] = fmaxf(a3 * inv + bl[lane + 96] + q3, 0.0f);
    }
    __syncthreads();

    // ---- GraphConv score: t[i]=h[i].Wrel, score[i]=h[i].Wroot+brel
    for (int i = tid; i < n; i += 256) {
      const float* Hr = Pb + (size_t)i * 256 + 128;
      float tt = 0.f, uu = 0.f;
      for (int c = 0; c < HDIM; c++) {
        const float hv = Hr[c];
        tt += hv * Wrel[c];
        uu += hv * Wroot[c];
      }
      s_t[i] = tt;
      s_score[i] = uu + brel;
    }
    __syncthreads();
    // score[i] += sum over masked incoming edges of t[src]
    for (int i = warp; i < n; i += 8) {
      float part = 0.f;
      for (int e = lane; e < NEDGE; e += 32)
        if (s_dst[e] == i && s_mask[e] > 0.5f) part += s_t[s_src[e]];
      for (int off = 16; off; off >>= 1) part += __shfl_down(part, off);
      if (lane == 0) s_score[i] += part;
    }
    __syncthreads();
    for (int i = tid; i < n; i += 256) s_work[i] = s_score[i];
    __syncthreads();

    // ---- top-k (max score, ties -> smaller index, like jax.lax.top_k)
    for (int r = 0; r < k; r++) {
      float bv = NEGINF;
      int   bi = 0x7fffffff;
      for (int i = tid; i < n; i += 256) {
        const float v = s_work[i];
        if (v > bv || (v == bv && i < bi)) { bv = v; bi = i; }
      }
      s_red[tid] = bv; s_redi[tid] = bi;
      __syncthreads();
      for (int off = 128; off; off >>= 1) {
        if (tid < off) {
          const float v2 = s_red[tid + off];
          const int   i2 = s_redi[tid + off];
          if (v2 > s_red[tid] || (v2 == s_red[tid] && i2 < s_redi[tid])) {
            s_red[tid] = v2; s_redi[tid] = i2;
          }
        }
        __syncthreads();
      }
      if (tid == 0) { s_perm[r] = s_redi[0]; s_work[s_redi[0]] = NEGINF; }
      __syncthreads();
    }

    // ---- node map + pooled x = h[perm] * tanh(score[perm])
    for (int i = tid; i < n; i += 256) s_nm[i] = -1;
    __syncthreads();
    for (int r = tid; r < k; r += 256) s_nm[s_perm[r]] = r;
    __syncthreads();
    for (int idx = tid; idx < k * HDIM; idx += 256) {
      const int r = idx >> 7, c = idx & 127;
      const int pi = s_perm[r];
      xbg[idx] = Pb[(size_t)pi * 256 + 128 + c] * tanhf(s_score[pi]);
    }
    __syncthreads();

    // ---- readout: accumulate [max ; mean] over pooled rows
    for (int c = tid; c < HDIM; c += 256) {
      float mx = NEGINF, sm = 0.f;
      for (int r = 0; r < k; r++) {
        const float v = xbg[r * HDIM + c];
        mx = fmaxf(mx, v);
        sm += v;
      }
      rg[c] += mx;
      rg[HDIM + c] += sm / (float)k;
    }
    __syncthreads();

    // ---- edge remap / mask filter
    for (int e = tid; e < NEDGE; e += 256) {
      const int ns = s_nm[s_src[e]];
      const int nd = s_nm[s_dst[e]];
      s_mask[e] *= (ns >= 0 && nd >= 0) ? 1.0f : 0.0f;
      s_src[e] = (ns >= 0) ? ns : 0;
      s_dst[e] = (nd >= 0) ? nd : 0;
    }

    // ---- next-layer projections: PQ2 = xnew @ [Wl;Wr]^T
    if (l < 2) {
      const float* Wl = (l == 0) ? p.Wl1 : p.Wl2;
      const float* Wr = (l == 0) ? p.Wr1 : p.Wr2;
      __syncthreads();
      for (int o = tid; o < k * 256; o += 256) {
        const int i = o >> 8, c = o & 255;
        const float* W = (c < 128) ? (Wl + (size_t)c * HDIM)
                                   : (Wr + (size_t)(c - 128) * HDIM);
        const float* xr = xbg + (size_t)i * HDIM;
        float acc = 0.f;
        for (int d = 0; d < HDIM; d++) acc += xr[d] * W[d];
        PQ2g[o] = acc;
      }
    }
    n = k;
    __syncthreads();
  }
}

// ---------------------------------------------------------------------------
// Kernel E: encoder MLP + gated attention pooling + heads (single block)
// ---------------------------------------------------------------------------
struct EP {
  const float* routs;    // [64,256]
  const int*   pat;      // [64]
  const float* W1; const float* b1;      // [128,256],[128]
  const float* W2; const float* b2;      // [32,128],[32]
  const float* Wa; const float* ba;      // [32,32],[32]
  const float* Wb; const float* bb;      // [32,32],[32]
  const float* Wc; const float* bc;      // [1,32],[1]
  const float* gW; const float* gb;      // [3,32],[3]
  const float* hW; const float* hb;      // [1,32],[1]
  float* out;                            // 512 + 48 + 16
};

__global__ __launch_bounds__(256) void gnn_head(EP p) {
  __shared__ float s_h1[BATCH * HDIM];   // 32 KB
  __shared__ float s_g[BATCH * FDIM];    // 8 KB
  __shared__ float s_A[BATCH];
  __shared__ float s_m[NPAT], s_s[NPAT];
  __shared__ float s_xp[NPAT * FDIM];
  const int tid = threadIdx.x;

  for (int o = tid; o < BATCH * HDIM; o += 256) {
    const int b = o >> 7, c = o & 127;
    const float* r = p.routs + (size_t)b * 256;
    const float* w = p.W1 + (size_t)c * 256;
    float acc = p.b1[c];
    for (int d = 0; d < 256; d++) acc += r[d] * w[d];
    s_h1[o] = fmaxf(acc, 0.0f);
  }
  __syncthreads();
  for (int o = tid; o < BATCH * FDIM; o += 256) {
    const int b = o >> 5, c = o & 31;
    const float* h = s_h1 + b * HDIM;
    const float* w = p.W2 + (size_t)c * HDIM;
    float acc = p.b2[c];
    for (int d = 0; d < HDIM; d++) acc += h[d] * w[d];
    s_g[o] = fmaxf(acc, 0.0f);
  }
  __syncthreads();
  if (tid < BATCH) {
    const float* gb = s_g + tid * FDIM;
    float acc = p.bc[0];
    for (int c = 0; c < FDIM; c++) {
      float av = p.ba[c], bv = p.bb[c];
      const float* wa = p.Wa + c * FDIM;
      const float* wb = p.Wb + c * FDIM;
      for (int d = 0; d < FDIM; d++) { av += gb[d] * wa[d]; bv += gb[d] * wb[d]; }
      acc += (1.0f / (1.0f + expf(-av))) * tanhf(bv) * p.Wc[c];
    }
    s_A[tid] = acc;
  }
  __syncthreads();
  if (tid < NPAT) {
    float m = NEGINF;
    for (int b = 0; b < BATCH; b++) if (p.pat[b] == tid) m = fmaxf(m, s_A[b]);
    s_m[tid] = m;
    float sm = 0.f;
    for (int b = 0; b < BATCH; b++) if (p.pat[b] == tid) sm += expf(s_A[b] - m);
    s_s[tid] = sm;
  }
  __syncthreads();
  for (int o = tid; o < NPAT * FDIM; o += 256) {
    const int pp = o >> 5, f = o & 31;
    float acc = 0.f;
    for (int b = 0; b < BATCH; b++)
      if (p.pat[b] == pp)
        acc += (expf(s_A[b] - s_m[pp]) / s_s[pp]) * s_g[b * FDIM + f];
    s_xp[o] = acc;
    p.out[o] = acc;                         // xp: [16,32]
  }
  __syncthreads();
  if (tid < NPAT) {
    const float* xr = s_xp + tid * FDIM;
    float lg[3];
    for (int c = 0; c < 3; c++) {
      float acc = p.gb[c];
      for (int d = 0; d < FDIM; d++) acc += xr[d] * p.gW[c * FDIM + d];
      lg[c] = acc;
    }
    const float m  = fmaxf(lg[0], fmaxf(lg[1], lg[2]));
    const float lse = m + logf(expf(lg[0] - m) + expf(lg[1] - m) + expf(lg[2] - m));
    for (int c = 0; c < 3; c++) p.out[NPAT * FDIM + tid * 3 + c] = lg[c] - lse;  // grade
    float hz = p.hb[0];
    for (int d = 0; d < FDIM; d++) hz += xr[d] * p.hW[d];
    p.out[NPAT * FDIM + NPAT * 3 + tid] = (1.0f / (1.0f + expf(-hz))) * 6.0f - 3.0f; // hazard
  }
}

// ---------------------------------------------------------------------------
extern "C" void kernel_launch(void* const* d_in, const int* in_sizes, int n_in,
                              void* d_out, int out_size, void* d_ws, size_t ws_size,
                              hipStream_t stream) {
  const float* x    = (const float*)d_in[0];
  const int*   ei   = (const int*)d_in[1];
  const int*   pat  = (const int*)d_in[2];
  // params in setup_inputs insertion order:
  const float* c0Wl = (const float*)d_in[3];
  const float* c0bl = (const float*)d_in[4];
  const float* c0Wr = (const float*)d_in[5];
  const float* p0Wrel = (const float*)d_in[6];
  const float* p0brel = (const float*)d_in[7];
  const float* p0Wroot= (const float*)d_in[8];
  const float* c1Wl = (const float*)d_in[9];
  const float* c1bl = (const float*)d_in[10];
  const float* c1Wr = (const float*)d_in[11];
  const float* p1Wrel = (const float*)d_in[12];
  const float* p1brel = (const float*)d_in[13];
  const float* p1Wroot= (const float*)d_in[14];
  const float* c2Wl = (const float*)d_in[15];
  const float* c2bl = (const float*)d_in[16];
  const float* c2Wr = (const float*)d_in[17];
  const float* p2Wrel = (const float*)d_in[18];
  const float* p2brel = (const float*)d_in[19];
  const float* p2Wroot= (const float*)d_in[20];
  const float* eW1 = (const float*)d_in[21];
  const float* eb1 = (const float*)d_in[22];
  const float* eW2 = (const float*)d_in[23];
  const float* eb2 = (const float*)d_in[24];
  const float* aWa = (const float*)d_in[25];
  const float* aba = (const float*)d_in[26];
  const float* aWb = (const float*)d_in[27];
  const float* abb = (const float*)d_in[28];
  const float* aWc = (const float*)d_in[29];
  const float* abc = (const float*)d_in[30];
  const float* gW  = (const float*)d_in[31];
  const float* gb  = (const float*)d_in[32];
  const float* hW  = (const float*)d_in[33];
  const float* hb  = (const float*)d_in[34];

  float* ws = (float*)d_ws;
  float* colinv = ws + OFF_COLINV;
  float* Wcat   = ws + OFF_WCAT;
  float* PQ     = ws + OFF_PQ;
  float* PQ2    = ws + OFF_PQ2;
  float* xb     = ws + OFF_XB;
  float* routs  = ws + OFF_ROUT;

  gnn_colmax<<<12, 256, 0, stream>>>(x, colinv);
  gnn_wcat<<<(256 * XDIM + 255) / 256, 256, 0, stream>>>(c0Wl, c0Wr, colinv, Wcat);
  // 64 graphs * 32 Mtiles * 4 strips = 8192 wave tasks / 8 waves per block
  gnn_gemm_wmma<<<1024, 256, 0, stream>>>(x, Wcat, PQ);

  DP dp;
  dp.ei = ei; dp.PQ = PQ; dp.PQ2 = PQ2; dp.xb = xb; dp.routs = routs;
  dp.bl0 = c0bl;
  dp.Wl1 = c1Wl; dp.bl1 = c1bl; dp.Wr1 = c1Wr;
  dp.Wl2 = c2Wl; dp.bl2 = c2bl; dp.Wr2 = c2Wr;
  dp.Wrel0 = p0Wrel; dp.brel0 = p0brel; dp.Wroot0 = p0Wroot;
  dp.Wrel1 = p1Wrel; dp.brel1 = p1brel; dp.Wroot1 = p1Wroot;
  dp.Wrel2 = p2Wrel; dp.brel2 = p2brel; dp.Wroot2 = p2Wroot;
  gnn_graph<<<BATCH, 256, 0, stream>>>(dp);

  EP ep;
  ep.routs = routs; ep.pat = pat;
  ep.W1 = eW1; ep.b1 = eb1; ep.W2 = eW2; ep.b2 = eb2;
  ep.Wa = aWa; ep.ba = aba; ep.Wb = aWb; ep.bb = abb; ep.Wc = aWc; ep.bc = abc;
  ep.gW = gW; ep.gb = gb; ep.hW = hW; ep.hb = hb;
  ep.out = (float*)d_out;
  gnn_head<<<1, 256, 0, stream>>>(ep);
}